// dynamic_conv_80857054314727
// MI455X (gfx1250) — compile-verified
//
#include <hip/hip_runtime.h>

// Dynamic grouped 7x7 convolution with per-pixel predicted weights (DDF-style).
//
// MI455X reasoning: ~2 GFLOP f32 vs ~38 MB mandatory HBM traffic -> memory/
// latency bound at 23.3 TB/s (~1.7us). Keep everything f32 and use
// V_WMMA_F32_16X16X4_F32 for the two 1x1-conv GEMMs (exact precision, CDNA5
// matrix path). Fully fused: per-16-pixel tile. The X[256x16] tile is DMA'd
// into LDS once per block by the Tensor Data Mover (tensor_load_to_lds with
// pad_interval=16dw/pad_amount=2dw -> conflict-free stride-18 rows), removing
// the 4x redundant per-wave global streams. The 784x16 dynamic-weight tile
// aliases the same LDS (barrier-separated lifetimes), ~61.5 KB total.

typedef __attribute__((ext_vector_type(2))) float v2f;
typedef __attribute__((ext_vector_type(8))) float v8f;
typedef __attribute__((ext_vector_type(4))) unsigned int v4u;
typedef __attribute__((ext_vector_type(8))) int v8i;
typedef __attribute__((ext_vector_type(4))) int v4i;

#define HWPIX 3136   // 56*56
#define WDIM  56
#define CIN   256
#define CMID  64
#define KTAPS 49     // 7*7
#define NGRP  16
#define GCH   16
#define NT    16     // pixel tile (one WMMA N dimension)
#define LDSTR 18     // LDS row stride in floats (no half-wave bank conflicts)

__global__ __launch_bounds__(128)
void dynconv_fused_kernel(const float* __restrict__ x, const float* __restrict__ y,
                          const float* __restrict__ w1, const float* __restrict__ gamma,
                          const float* __restrict__ beta, const float* __restrict__ mean,
                          const float* __restrict__ var, const float* __restrict__ w2,
                          const float* __restrict__ b2, float* __restrict__ out)
{
    // smem serves two barrier-separated lives:
    //   stage 1: X tile   [256][16] stride 18 (first 18 KB), filled by TDM
    //   stage 2/3: weight tile [784][16] stride 18 (56.4 KB)
    __shared__ float smem[(KTAPS * NGRP) * LDSTR];
    __shared__ float ts[CMID * LDSTR];          // T tile [64][16] (4.5 KB)
    __shared__ float ssc[CMID];                 // BN scale
    __shared__ float ssh[CMID];                 // BN shift

    float* xs = smem;                           // alias: X tile (stage 1 only)
    float* wl = smem;                           // alias: weight tile (stage 2+)

    const int bi   = blockIdx.x;
    const int b    = bi / (HWPIX / NT);
    const int hw0  = (bi % (HWPIX / NT)) * NT;

    const int tid  = threadIdx.x;
    const int wave = tid >> 5;      // 0..3 (wave32)
    const int lane = tid & 31;
    const int half = lane >> 4;     // half-wave select
    const int lm   = lane & 15;     // m (A rows) / n (B,D cols)

    // ---- TDM: async-load X[256 ch][16 px] tile into LDS (one DMA per block).
    // D# per CDNA5 ISA 8.3/8.4: 2D tile, data_size=4B, pad 2 dwords every 16
    // dwords -> LDS row stride 18 floats. Groups 2/3 (and the extra group in
    // the 6-arg builtin form) unused -> zero for a 2D tile.
    if (wave == 0) {
        unsigned long long gaddr =
            (unsigned long long)(uintptr_t)(x + (size_t)b * CIN * HWPIX + hw0);
        unsigned lds_off = (unsigned)(uintptr_t)xs;   // LDS aperture: low 32 bits = byte offset

        v4u g0;
        g0[0] = 1u;                                          // count=1 (valid user D#)
        g0[1] = lds_off;                                     // lds_addr
        g0[2] = (unsigned)(gaddr & 0xffffffffu);             // global_addr[31:0]
        g0[3] = (unsigned)((gaddr >> 32) & 0x01ffffffu)      // global_addr[56:32]
              | (2u << 30);                                  // type=2 ("image")

        v8i g1;
        g1[0] = (int)((2u << 16)      // data_size: 4 bytes
                    | (1u << 20)      // pad_enable
                    | (3u << 22)      // pad_interval: 16 dwords
                    | (1u << 25));    // pad_amount: 2 dwords
        g1[1] = (int)((HWPIX & 0xffffu) << 16);  // tensor_dim0[15:0] (3136)
        g1[2] = (int)((unsigned)CIN << 16);      // tensor_dim0 hi=0 | tensor_dim1 lo (256)
        g1[3] = (int)((unsigned)NT << 16);       // tensor_dim1 hi=0 | tile_dim0 (16)
        g1[4] = (int)CIN;                        // tile_dim1 (256) | tile_dim2=0
        g1[5] = (int)HWPIX;                      // tensor_dim0_stride[31:0] (3136)
        g1[6] = 0;                               // stride0 hi | tensor_dim1_stride lo
        g1[7] = 0;

        v4i gz4 = {0, 0, 0, 0};
        v8i gz8 = {0, 0, 0, 0, 0, 0, 0, 0};
        __builtin_amdgcn_tensor_load_to_lds(g0, g1, gz4, gz4, gz8, 0);
    }

    // Precompute BN scale/shift once per block (overlaps the DMA).
    if (tid < CMID) {
        float s = gamma[tid] * rsqrtf(var[tid] + 1e-5f);
        ssc[tid] = s;
        ssh[tid] = beta[tid] - mean[tid] * s;
    }
    if (wave == 0) __builtin_amdgcn_s_wait_tensorcnt(0);
    __syncthreads();

    // ---------------- Stage 1: T[64x16] = relu(bn(w1[64x256] @ X[256x16]))
    // Each wave owns one 16-row M tile. K loop: 256 in steps of 4.
    {
        const int mt = wave;
        v8f acc = {0.f,0.f,0.f,0.f,0.f,0.f,0.f,0.f};
        // A: 16x4 f32 layout -> lane holds A[m][k+2*half], A[m][k+2*half+1]
        const float* ap = w1 + (mt * 16 + lm) * CIN + 2 * half;
        // B: 4x16 f32 layout (dual of A) -> lane holds X[k+2*half][n], X[k+2*half+1][n]
        const float* bp = xs + (2 * half) * LDSTR + lm;
        #pragma unroll 4
        for (int k = 0; k < CIN; k += 4) {
            v2f a = *(const v2f*)(ap + k);               // 8B-aligned global float2
            v2f bb;
            bb.x = bp[k * LDSTR];
            bb.y = bp[(k + 1) * LDSTR];
            acc = __builtin_amdgcn_wmma_f32_16x16x4_f32(false, a, false, bb,
                                                        (short)0, acc, false, false);
        }
        // D element v -> (m = v + 8*half, n = lm); apply BN + ReLU, park in LDS.
        #pragma unroll
        for (int v = 0; v < 8; ++v) {
            int ch = mt * 16 + v + 8 * half;
            float t = acc[v] * ssc[ch] + ssh[ch];
            ts[ch * LDSTR + lm] = fmaxf(t, 0.f);
        }
    }
    __syncthreads();   // also ends the X-tile lifetime of smem

    // ---------------- Stage 2: W[784x16] = w2[784x64] @ T[64x16] + b2
    // 49 M tiles round-robined over the 4 waves. K loop: 64 in steps of 4.
    for (int mt = wave; mt < KTAPS; mt += 4) {
        v8f acc = {0.f,0.f,0.f,0.f,0.f,0.f,0.f,0.f};
        const float* ap = w2 + (mt * 16 + lm) * CMID + 2 * half;
        const float* tp = ts + (2 * half) * LDSTR + lm;
        #pragma unroll 4
        for (int k = 0; k < CMID; k += 4) {
            v2f a = *(const v2f*)(ap + k);
            v2f bb;
            bb.x = tp[k * LDSTR];
            bb.y = tp[(k + 1) * LDSTR];
            acc = __builtin_amdgcn_wmma_f32_16x16x4_f32(false, a, false, bb,
                                                        (short)0, acc, false, false);
        }
        #pragma unroll
        for (int v = 0; v < 8; ++v) {
            int o = mt * 16 + v + 8 * half;
            wl[o * LDSTR + lm] = acc[v] + b2[o];
        }
    }
    __syncthreads();

    // ---------------- Stage 3: dynamic 7x7 grouped aggregation over y.
    // 256 (group, pixel) pairs -> 2 per thread; 16 channels per pair.
    for (int pass = 0; pass < 2; ++pass) {
        int idx = tid + pass * 128;          // 0..255
        int g = idx >> 4;
        int n = idx & 15;
        int p = hw0 + n;
        int h = p / WDIM;
        int w = p % WDIM;

        float acc[GCH];
        #pragma unroll
        for (int c = 0; c < GCH; ++c) acc[c] = 0.f;

        const float* yg = y + ((size_t)b * CIN + g * GCH) * HWPIX;
        const float* wr = wl + (g * KTAPS) * LDSTR + n;

        for (int kk = 0; kk < KTAPS; ++kk) {
            int di = kk / 7 - 3;
            int dj = kk - (kk / 7) * 7 - 3;
            int hh = h + di;
            int ww = w + dj;
            if (hh >= 0 && hh < WDIM && ww >= 0 && ww < WDIM) {  // zero padding
                float wt = wr[kk * LDSTR];
                const float* yp = yg + hh * WDIM + ww;
                #pragma unroll
                for (int c = 0; c < GCH; ++c) acc[c] += wt * yp[(size_t)c * HWPIX];
            }
        }
        float* op = out + ((size_t)b * CIN + g * GCH) * HWPIX + p;
        #pragma unroll
        for (int c = 0; c < GCH; ++c) op[(size_t)c * HWPIX] = acc[c];
    }
}

extern "C" void kernel_launch(void* const* d_in, const int* in_sizes, int n_in,
                              void* d_out, int out_size, void* d_ws, size_t ws_size,
                              hipStream_t stream) {
    const float* x     = (const float*)d_in[0];
    const float* y     = (const float*)d_in[1];
    const float* w1    = (const float*)d_in[2];
    const float* gamma = (const float*)d_in[3];
    const float* beta  = (const float*)d_in[4];
    const float* mean  = (const float*)d_in[5];
    const float* var   = (const float*)d_in[6];
    const float* w2    = (const float*)d_in[7];
    const float* b2    = (const float*)d_in[8];
    float* out = (float*)d_out;

    const int nblocks = 4 * (HWPIX / NT);   // B * pixel-tiles = 784
    dynconv_fused_kernel<<<nblocks, 128, 0, stream>>>(x, y, w1, gamma, beta,
                                                      mean, var, w2, b2, out);
}